// CriticUp_71665824301172
// MI455X (gfx1250) — compile-verified
//
#include <hip/hip_runtime.h>
#include <cstddef>

typedef float v2f __attribute__((ext_vector_type(2)));
typedef float v8f __attribute__((ext_vector_type(8)));

#define WAVES_PER_BLOCK 8
#define BLOCK_THREADS   256

// Problem dims (from reference)
#define S_DIM  64
#define A_DIM  16
#define MSG_D  32
#define C_D    4
#define CH_D   73      // 9 + S
#define EPSF   1e-12f

// ---- LDS layout (float offsets) ----
// Weights stored K-pair interleaved: for even k, (Wt[k][n], Wt[k+1][n]) are adjacent,
// so a WMMA B-fragment is ONE 8B-aligned ds_load_b64 into an even VGPR pair.
// Pair-row strides chosen so 2*stride == 32 (mod 64): the two 16-lane halves
// (reading pair rows 2s and 2s+1) hit disjoint LDS bank halves.
#define W1P_STRIDE 80   // W1p[40 pairs][64] (K=80)
#define W2P_STRIDE 80   // W2p[48 pairs][64] (K=96)
#define W3P_STRIDE 48   // W3p[32 pairs][32] (K=64)
#define WAP_STRIDE 48   // Wap[38 pairs][32] (K=73 zero-padded to 76)
#define OFF_W1P 0
#define OFF_W2P (OFF_W1P + 40*W1P_STRIDE*2)
#define OFF_W3P (OFF_W2P + 48*W2P_STRIDE*2)
#define OFF_WAP (OFF_W3P + 32*W3P_STRIDE*2)
#define OFF_B1  (OFF_WAP + 38*WAP_STRIDE*2)
#define OFF_B2  (OFF_B1 + 64)
#define OFF_B3  (OFF_B2 + 64)
#define OFF_BA  (OFF_B3 + 32)
#define OFF_BUF (OFF_BA + 32)
#define BUF_STRIDE 100              // 16-row A-staging tile, padded stride (conflict-free)
#define BUF_PER_WAVE (16*BUF_STRIDE)
#define LDS_FLOATS (OFF_BUF + WAVES_PER_BLOCK*BUF_PER_WAVE)

// pair-layout address for weight element (k, o): ((k/2)*stride + o)*2 + (k&1)
#define WPAIR(off, strd, k, o) ((off) + ((((k) >> 1) * (strd) + (o)) << 1) + ((k) & 1))

static __device__ __forceinline__ v8f v8_bcast(float v) {
    v8f r;
#pragma unroll
    for (int i = 0; i < 8; ++i) r[i] = v;
    return r;
}

// D = A(16x4 f32) * B(4x16 f32) + C  -- full fp32 matrix op (matches reference precision)
static __device__ __forceinline__ v8f wmma4(v2f a, v2f b, v8f c) {
    return __builtin_amdgcn_wmma_f32_16x16x4_f32(false, a, false, b, (short)0, c, false, false);
}

__global__ __launch_bounds__(BLOCK_THREADS, 1)
void critic_up_kernel(const float* __restrict__ x,  const float* __restrict__ u,
                      const float* __restrict__ cs, const float* __restrict__ m,
                      const float* __restrict__ W1, const float* __restrict__ b1,
                      const float* __restrict__ W2, const float* __restrict__ b2,
                      const float* __restrict__ W3, const float* __restrict__ b3,
                      const float* __restrict__ Wa, const float* __restrict__ ba,
                      float* __restrict__ out, int nTiles)
{
    extern __shared__ float lds[];
    const int tid = threadIdx.x;

    // ---- one-time per block: load + transpose weights into pair-interleaved LDS ----
    for (int i = tid; i < 64*80; i += BLOCK_THREADS) {   // W1 [64][80]
        int o = i / 80, k = i % 80;
        lds[WPAIR(OFF_W1P, W1P_STRIDE, k, o)] = W1[i];
    }
    for (int i = tid; i < 64*96; i += BLOCK_THREADS) {   // W2 [64][96]
        int o = i / 96, k = i % 96;
        lds[WPAIR(OFF_W2P, W2P_STRIDE, k, o)] = W2[i];
    }
    for (int i = tid; i < 32*64; i += BLOCK_THREADS) {   // W3 [32][64]
        int o = i / 64, k = i % 64;
        lds[WPAIR(OFF_W3P, W3P_STRIDE, k, o)] = W3[i];
    }
    for (int i = tid; i < 32*73; i += BLOCK_THREADS) {   // Wa [32][73]
        int o = i / 73, k = i % 73;
        lds[WPAIR(OFF_WAP, WAP_STRIDE, k, o)] = Wa[i];
    }
    for (int i = tid; i < 3*32; i += BLOCK_THREADS) {    // zero K-pad rows 73..75 of Wa
        int k = 73 + i / 32, o = i % 32;
        lds[WPAIR(OFF_WAP, WAP_STRIDE, k, o)] = 0.0f;
    }
    if (tid < 64) { lds[OFF_B1 + tid] = b1[tid]; lds[OFF_B2 + tid] = b2[tid]; }
    if (tid < 32) { lds[OFF_B3 + tid] = b3[tid]; lds[OFF_BA + tid] = ba[tid]; }
    __syncthreads();

    const int wave = tid >> 5;
    const int lane = tid & 31;
    const int g    = lane >> 4;   // lane-half (selects K pair / M high rows per WMMA layout)
    const int ln   = lane & 15;   // lane within half
    float* buf = &lds[OFF_BUF + wave * BUF_PER_WAVE];   // wave-private staging: no barriers

    const int stride = gridDim.x * WAVES_PER_BLOCK;
    for (int tIdx = blockIdx.x * WAVES_PER_BLOCK + wave; tIdx < nTiles; tIdx += stride) {
        const int r0 = tIdx * 16;

        // prefetch next tile's children block (dominant input stream)
        {
            int nxt = tIdx + stride;
            if (nxt < nTiles) {
                const char* p = (const char*)&cs[(size_t)nxt * 16 * (C_D * CH_D)];
                __builtin_prefetch(p + lane * 64, 0, 1);
            }
        }

        // ---------------- attention logits: [16 rows] x [4 children] -> [16][32] each -----
        v8f lg[C_D][2];
        // zero A-tile pad cols 73..75 (16 rows x 3 cols)
        for (int idx = lane; idx < 48; idx += 32)
            buf[(idx / 3) * BUF_STRIDE + 73 + (idx % 3)] = 0.0f;
#pragma unroll
        for (int c = 0; c < C_D; ++c) {
            // stage children_states[r0..r0+15][c][0..72] coalesced into buf[r][k]
            for (int idx = lane; idx < 16 * CH_D; idx += 32) {
                int r = idx / CH_D, k = idx % CH_D;
                buf[r * BUF_STRIDE + k] = cs[(size_t)(r0 + r) * (C_D * CH_D) + c * CH_D + k];
            }
#pragma unroll
            for (int t = 0; t < 2; ++t) {
                v8f acc = v8_bcast(lds[OFF_BA + 16 * t + ln]);
                for (int s = 0; s < 19; ++s) {            // K = 76 (73 + zero pad)
                    int k = 4 * s + 2 * g;                // even
                    v2f a = *(const v2f*)&buf[ln * BUF_STRIDE + k];
                    v2f b = *(const v2f*)&lds[WPAIR(OFF_WAP, WAP_STRIDE, k, 16 * t + ln)];
                    acc = wmma4(a, b, acc);
                }
                lg[c][t] = acc;
            }
        }

        // -------- softmax over children (pure per-lane: same (m,n) slot in all 4 accs) ----
        v8f magg[2];
#pragma unroll
        for (int t = 0; t < 2; ++t) {
            v8f mg;
#pragma unroll
            for (int j = 0; j < 8; ++j) {
                float l0 = lg[0][t][j], l1 = lg[1][t][j], l2 = lg[2][t][j], l3 = lg[3][t][j];
                float mx = fmaxf(fmaxf(l0, l1), fmaxf(l2, l3));
                float e0 = __expf(l0 - mx), e1 = __expf(l1 - mx);
                float e2 = __expf(l2 - mx), e3 = __expf(l3 - mx);
                float inv = 1.0f / (e0 + e1 + e2 + e3);
                size_t rowBase = (size_t)(r0 + j + 8 * g) * (C_D * MSG_D);
                int col = 16 * t + ln;
                float m0 = m[rowBase + 0 * MSG_D + col];
                float m1 = m[rowBase + 1 * MSG_D + col];
                float m2 = m[rowBase + 2 * MSG_D + col];
                float m3 = m[rowBase + 3 * MSG_D + col];
                mg[j] = (m0 * e0 + m1 * e1 + m2 * e2 + m3 * e3) * inv;
            }
            magg[t] = mg;
        }

        // ---------------- fc1: [16][80] @ W1 -> [16][64], then L2-normalize rows ----------
        {
            const float4* xv = (const float4*)(x + (size_t)r0 * S_DIM);
            for (int idx = lane; idx < 16 * 16; idx += 32) {           // 16 rows x 16 float4
                int r = idx >> 4, c4 = idx & 15;
                *(float4*)&buf[r * BUF_STRIDE + c4 * 4] = xv[r * 16 + c4];
            }
            const float4* uv = (const float4*)(u + (size_t)r0 * A_DIM);
            for (int idx = lane; idx < 16 * 4; idx += 32) {            // 16 rows x 4 float4
                int r = idx >> 2, c4 = idx & 3;
                *(float4*)&buf[r * BUF_STRIDE + 64 + c4 * 4] = uv[r * 4 + c4];
            }
        }
        v8f xu[4];
#pragma unroll
        for (int t = 0; t < 4; ++t) {
            v8f acc = v8_bcast(lds[OFF_B1 + 16 * t + ln]);
            for (int s = 0; s < 20; ++s) {                 // K = 80
                int k = 4 * s + 2 * g;
                v2f a = *(const v2f*)&buf[ln * BUF_STRIDE + k];
                v2f b = *(const v2f*)&lds[WPAIR(OFF_W1P, W1P_STRIDE, k, 16 * t + ln)];
                acc = wmma4(a, b, acc);
            }
            xu[t] = acc;
        }
        // row L2 norm: per-lane partial over 4 N-tiles, then 16-lane tree (stays in half)
#pragma unroll
        for (int j = 0; j < 8; ++j) {
            float s = xu[0][j]*xu[0][j] + xu[1][j]*xu[1][j] + xu[2][j]*xu[2][j] + xu[3][j]*xu[3][j];
            s += __shfl_xor(s, 1, 32);
            s += __shfl_xor(s, 2, 32);
            s += __shfl_xor(s, 4, 32);
            s += __shfl_xor(s, 8, 32);
            float scale = 1.0f / fmaxf(sqrtf(s), EPSF);
            int row = j + 8 * g;
#pragma unroll
            for (int t = 0; t < 4; ++t)
                buf[row * BUF_STRIDE + 16 * t + ln] = tanhf(xu[t][j] * scale);   // xum[0:64]
            buf[row * BUF_STRIDE + 64 + ln] = tanhf(magg[0][j]);                 // xum[64:80]
            buf[row * BUF_STRIDE + 80 + ln] = tanhf(magg[1][j]);                 // xum[80:96]
        }

        // ---------------- fc2: [16][96] @ W2 -> tanh -> [16][64] --------------------------
        v8f h[4];
#pragma unroll
        for (int t = 0; t < 4; ++t) {
            v8f acc = v8_bcast(lds[OFF_B2 + 16 * t + ln]);
            for (int s = 0; s < 24; ++s) {                 // K = 96
                int k = 4 * s + 2 * g;
                v2f a = *(const v2f*)&buf[ln * BUF_STRIDE + k];
                v2f b = *(const v2f*)&lds[WPAIR(OFF_W2P, W2P_STRIDE, k, 16 * t + ln)];
                acc = wmma4(a, b, acc);
            }
#pragma unroll
            for (int j = 0; j < 8; ++j) acc[j] = tanhf(acc[j]);
            h[t] = acc;
        }
#pragma unroll
        for (int t = 0; t < 4; ++t)
#pragma unroll
            for (int j = 0; j < 8; ++j)
                buf[(j + 8 * g) * BUF_STRIDE + 16 * t + ln] = h[t][j];

        // ---------------- fc3: [16][64] @ W3 -> [16][32], L2-normalize, store -------------
        v8f o[2];
#pragma unroll
        for (int t = 0; t < 2; ++t) {
            v8f acc = v8_bcast(lds[OFF_B3 + 16 * t + ln]);
            for (int s = 0; s < 16; ++s) {                 // K = 64
                int k = 4 * s + 2 * g;
                v2f a = *(const v2f*)&buf[ln * BUF_STRIDE + k];
                v2f b = *(const v2f*)&lds[WPAIR(OFF_W3P, W3P_STRIDE, k, 16 * t + ln)];
                acc = wmma4(a, b, acc);
            }
            o[t] = acc;
        }
#pragma unroll
        for (int j = 0; j < 8; ++j) {
            float s = o[0][j]*o[0][j] + o[1][j]*o[1][j];
            s += __shfl_xor(s, 1, 32);
            s += __shfl_xor(s, 2, 32);
            s += __shfl_xor(s, 4, 32);
            s += __shfl_xor(s, 8, 32);
            float scale = 1.0f / fmaxf(sqrtf(s), EPSF);
            size_t row = (size_t)(r0 + j + 8 * g);
            out[row * MSG_D + ln]      = o[0][j] * scale;
            out[row * MSG_D + 16 + ln] = o[1][j] * scale;
        }
    }
}

extern "C" void kernel_launch(void* const* d_in, const int* in_sizes, int n_in,
                              void* d_out, int out_size, void* d_ws, size_t ws_size,
                              hipStream_t stream) {
    const float* x  = (const float*)d_in[0];
    const float* u  = (const float*)d_in[1];
    const float* cs = (const float*)d_in[2];
    const float* m  = (const float*)d_in[3];
    const float* W1 = (const float*)d_in[4];
    const float* b1 = (const float*)d_in[5];
    const float* W2 = (const float*)d_in[6];
    const float* b2 = (const float*)d_in[7];
    const float* W3 = (const float*)d_in[8];
    const float* b3 = (const float*)d_in[9];
    const float* Wa = (const float*)d_in[10];
    const float* ba = (const float*)d_in[11];
    float* out = (float*)d_out;

    const int B = in_sizes[0] / S_DIM;          // 524288
    const int nTiles = B / 16;                  // 32768 16-row tiles
    int blocks = (nTiles + WAVES_PER_BLOCK - 1) / WAVES_PER_BLOCK;
    if (blocks > 2048) blocks = 2048;           // grid-stride amortizes weight staging
    const size_t shmem = (size_t)LDS_FLOATS * sizeof(float);   // ~132 KB -> 2 blocks/WGP

    hipLaunchKernelGGL(critic_up_kernel, dim3(blocks), dim3(BLOCK_THREADS), shmem, stream,
                       x, u, cs, m, W1, b1, W2, b2, W3, b3, Wa, ba, out, nTiles);
}